// TemporalSampling_62534723829931
// MI455X (gfx1250) — compile-verified
//
#include <hip/hip_runtime.h>

typedef float v2f __attribute__((ext_vector_type(2)));
typedef float v8f __attribute__((ext_vector_type(8)));

#define CCH 64
#define HH 100
#define WW 252
#define HWSZ (HH * WW)          // 25200, divisible by 16
#define NBMAX 4                 // reference record_len max 5 -> max 4 neighbors
#define TILES_PER_GROUP (HWSZ / 16)

// ---------------------------------------------------------------------------
// Per-wave (16 pixels, one group) attention score -> 0/1 mask.
//   logits_l = diag( Ego(16x64) x Nb_l(64x16) )  via chained V_WMMA_F32_16X16X4_F32
//   q_l      = w . nb_l  accumulated in VALU from the B fragments directly
//              (lane halves own disjoint channel sets -> shfl_xor(16) combine)
//   score    = softmax_l(logits/8) . q  + b ;  mask = score > 0
// NL is compile-time so the inner loop is straight-line (no uniform-branch
// chains between loads and WMMAs; EXEC stays all-ones around every WMMA).
// ---------------------------------------------------------------------------
template <int NL>
__device__ __forceinline__ void score_tile(
    const float* __restrict__ ego, const float* __restrict__ mlp_w,
    float b0, int lane, float* __restrict__ dst)
{
    const int half = lane >> 4;                 // K-pair select per ISA layout
    const int lm   = lane & 15;                 // M (A rows) / N (B cols) index
    const size_t chw = (size_t)CCH * HWSZ;

    v8f   dlog[NL];
    float qacc[NL];
#pragma unroll
    for (int l = 0; l < NL; ++l) { dlog[l] = (v8f){}; qacc[l] = 0.0f; }

    for (int s = 0; s < CCH / 4; ++s) {
        const int c0 = 4 * s + 2 * half;
        v2f a;
        a.x = ego[(size_t)(c0 + 0) * HWSZ + lm];    // A: row M=lm, K=c0..c0+1
        a.y = ego[(size_t)(c0 + 1) * HWSZ + lm];
        const float w0 = mlp_w[c0 + 0];             // broadcast (L2/const cached)
        const float w1 = mlp_w[c0 + 1];
#pragma unroll
        for (int l = 0; l < NL; ++l) {
            const float* nb = ego + (size_t)(l + 1) * chw;
            v2f b;
            b.x = nb[(size_t)(c0 + 0) * HWSZ + lm]; // B: col N=lm, K=c0..c0+1
            b.y = nb[(size_t)(c0 + 1) * HWSZ + lm];
            dlog[l] = __builtin_amdgcn_wmma_f32_16x16x4_f32(
                false, a, false, b, (short)0, dlog[l], false, false);
            qacc[l] = fmaf(w0, b.x, qacc[l]);       // partial w . nb_l per lane
            qacc[l] = fmaf(w1, b.y, qacc[l]);
        }
    }

    // Extract diag(D): D[m,n] lives at vgpr m%8, lane n + 16*(m>=8).
    // Diag (r,r)@(vgpr r, lane r); (r+8,r+8)@(vgpr r, lane r+24).
    float lg[NL], qv[NL];
#pragma unroll
    for (int l = 0; l < NL; ++l) {
        float acc = 0.0f;
#pragma unroll
        for (int r = 0; r < 8; ++r) {
            float vr = dlog[l][r];
            float d0 = __shfl(vr, r, 32);
            float d1 = __shfl(vr, r + 24, 32);
            acc = (lane == r)     ? d0 : acc;
            acc = (lane == r + 8) ? d1 : acc;
        }
        lg[l] = acc * 0.125f;                       // / sqrt(C=64)
        // combine the two lane-half channel partial sums: q_l[n=lm]
        qv[l] = qacc[l] + __shfl_xor(qacc[l], 16, 32);
    }

    if (lane < 16) {
        float mx = -3.4e38f;
#pragma unroll
        for (int l = 0; l < NL; ++l) mx = fmaxf(mx, lg[l]);
        float se = 0.0f, num = 0.0f;
#pragma unroll
        for (int l = 0; l < NL; ++l) {
            float e = __expf(lg[l] - mx);
            se  += e;
            num += e * qv[l];
        }
        float score = num / se + b0;                // sigmoid(x)>0.5 <=> x>0
        dst[lane] = (score > 0.0f) ? 1.0f : 0.0f;
    }
}

__global__ __launch_bounds__(256) void score_mask_kernel(
    const float* __restrict__ feats, const float* __restrict__ mlp_w,
    const float* __restrict__ mlp_b, const int* __restrict__ rl,
    float* __restrict__ ws_mask, int ngroups)
{
    const int lane = threadIdx.x & 31;
    const int wid  = blockIdx.x * (blockDim.x >> 5) + (threadIdx.x >> 5);
    const int total = ngroups * TILES_PER_GROUP;
    if (wid >= total) return;                       // wave-uniform

    const int g  = wid / TILES_PER_GROUP;
    const int n0 = (wid - g * TILES_PER_GROUP) * 16;

    int img0 = 0;
    for (int j = 0; j < g; ++j) img0 += rl[j];
    int nl = rl[g] - 1;
    // force wave-uniformity: scalar branches only, EXEC stays all-1 for WMMA
    nl   = __builtin_amdgcn_readfirstlane(nl);
    img0 = __builtin_amdgcn_readfirstlane(img0);

    float* dst = ws_mask + (size_t)g * HWSZ + n0;
    if (nl <= 0) {                                  // single-image group
        if (lane < 16) dst[lane] = 1.0f;
        return;
    }

    const float b0 = mlp_b[0];
    const float* ego = feats + (size_t)img0 * ((size_t)CCH * HWSZ) + n0;

    if (nl >= 4)      score_tile<4>(ego, mlp_w, b0, lane, dst);
    else if (nl == 3) score_tile<3>(ego, mlp_w, b0, lane, dst);
    else if (nl == 2) score_tile<2>(ego, mlp_w, b0, lane, dst);
    else              score_tile<1>(ego, mlp_w, b0, lane, dst);
}

// ---------------------------------------------------------------------------
// Assemble output. Ego rows = 1.0; neighbor rows = 3x3 max dilation of the
// group's shared mask (SAME padding via bounded window; values are 0/1).
// ---------------------------------------------------------------------------
__global__ __launch_bounds__(256) void dilate_kernel(
    const float* __restrict__ ws_mask, const int* __restrict__ rl,
    float* __restrict__ out, int ngroups, int nimg)
{
    int idx = blockIdx.x * blockDim.x + threadIdx.x;
    if (idx >= nimg * HWSZ) return;
    int i = idx / HWSZ;
    int p = idx - i * HWSZ;
    int h = p / WW, w = p - h * WW;

    int acc = 0, g = 0, ego = 0;
    for (int j = 0; j < ngroups; ++j) {
        int r = rl[j];
        if (i >= acc && i < acc + r) { g = j; ego = (i == acc); }
        acc += r;
    }

    float v;
    if (ego) {
        v = 1.0f;
    } else {
        v = 0.0f;
        const float* m = ws_mask + (size_t)g * HWSZ;
        for (int dh = -1; dh <= 1; ++dh) {
            int hh = h + dh;
            if (hh < 0 || hh >= HH) continue;
            for (int dw = -1; dw <= 1; ++dw) {
                int wv = w + dw;
                if (wv < 0 || wv >= WW) continue;
                v = fmaxf(v, m[hh * WW + wv]);
            }
        }
    }
    out[idx] = v;
}

extern "C" void kernel_launch(void* const* d_in, const int* in_sizes, int n_in,
                              void* d_out, int out_size, void* d_ws, size_t ws_size,
                              hipStream_t stream)
{
    const float* feats = (const float*)d_in[0];
    const float* mlp_w = (const float*)d_in[1];
    const float* mlp_b = (const float*)d_in[2];
    const int*   rl    = (const int*)d_in[3];

    const int ngroups = in_sizes[3];
    const int nimg    = in_sizes[0] / (CCH * HWSZ);
    float* ws_mask    = (float*)d_ws;   // ngroups * HWSZ floats

    const int totalWaves = ngroups * TILES_PER_GROUP;
    const int wavesPerBlock = 8;        // 256 threads = 8 wave32
    const int blocks1 = (totalWaves + wavesPerBlock - 1) / wavesPerBlock;
    score_mask_kernel<<<blocks1, 256, 0, stream>>>(feats, mlp_w, mlp_b, rl,
                                                   ws_mask, ngroups);

    const int totalElems = nimg * HWSZ;
    const int blocks2 = (totalElems + 255) / 256;
    dilate_kernel<<<blocks2, 256, 0, stream>>>(ws_mask, rl, (float*)d_out,
                                               ngroups, nimg);
}